// Transformer_18597208391716
// MI455X (gfx1250) — compile-verified
//
#include <hip/hip_runtime.h>
#include <hip/hip_bf16.h>

typedef __attribute__((ext_vector_type(2))) float v2f;
typedef __attribute__((ext_vector_type(8))) float v8f;

__device__ __forceinline__ v8f wmma4(v2f a, v2f b, v8f c) {
    // V_WMMA_F32_16X16X4_F32 : D = A(16x4) * B(4x16) + C(16x16), fp32 throughout
    return __builtin_amdgcn_wmma_f32_16x16x4_f32(false, a, false, b, (short)0, c, false, false);
}

// -------------------------------------------------------------------------
// mask -> float {0,1}
// -------------------------------------------------------------------------
__global__ void mf_kernel(const int* __restrict__ mask, float* __restrict__ mfv, int n) {
    int i = blockIdx.x * blockDim.x + threadIdx.x;
    if (i < n) mfv[i] = mask[i] ? 1.0f : 0.0f;
}

// -------------------------------------------------------------------------
// LayerNorm over last dim (D=256), one block (256 threads) per row
// -------------------------------------------------------------------------
__global__ void ln_kernel(const float* __restrict__ x, float* __restrict__ out,
                          const float* __restrict__ g, const float* __restrict__ b) {
    __shared__ float red[256];
    const int row = blockIdx.x;
    const int tid = threadIdx.x;
    float v = x[(size_t)row * 256 + tid];
    red[tid] = v;
    __syncthreads();
    for (int s = 128; s > 0; s >>= 1) {
        if (tid < s) red[tid] += red[tid + s];
        __syncthreads();
    }
    float mean = red[0] * (1.0f / 256.0f);
    __syncthreads();
    float d = v - mean;
    red[tid] = d * d;
    __syncthreads();
    for (int s = 128; s > 0; s >>= 1) {
        if (tid < s) red[tid] += red[tid + s];
        __syncthreads();
    }
    float var = red[0] * (1.0f / 256.0f);
    float r = rsqrtf(var + 1e-6f);
    out[(size_t)row * 256 + tid] = d * r * g[tid] + b[tid];
}

// -------------------------------------------------------------------------
// Fused GEMM:  C[M,N] = epilogue(A[M,K] @ W[K,N])
// epilogue: (+bias[n]) -> (relu) -> (*mf[m]) -> (+res[m,n])
// Tile: 64(M) x 128(N) per block, 256 threads = 8 waves, wave tile 32x32.
// A tile row-contiguous [64][34]; B tile pair-interleaved [k/2][n][k&1] so
// every WMMA operand is one aligned ds_load_b64, bank-conflict free.
// grid = (M/64, N/128)
// -------------------------------------------------------------------------
__global__ void gemm_kernel(const float* __restrict__ A, const float* __restrict__ W,
                            float* __restrict__ C, const float* __restrict__ bias,
                            const float* __restrict__ mfv, const float* __restrict__ res,
                            int K, int N, int relu) {
    __shared__ __align__(16) float As[64][34];        // stride 34 -> A-frag reads hit all 64 banks
    __shared__ __align__(16) float Bs[16][144][2];    // kp stride 288 dwords == 32 mod 64

    const int tid = threadIdx.x;
    const int lane = tid & 31;
    const int wave = tid >> 5;
    const int half = lane >> 4;
    const int lane16 = lane & 15;
    const int wm = wave >> 2;
    const int wn = wave & 3;
    const int m0 = blockIdx.x * 64;
    const int n0 = blockIdx.y * 128;

    const int rowA0 = wm * 32 + lane16;
    const int rowA1 = rowA0 + 16;
    const int colB0 = wn * 32 + lane16;
    const int colB1 = colB0 + 16;

    v8f acc[2][2] = {};

    for (int k0 = 0; k0 < K; k0 += 32) {
        // A tile: coalesced rows
        for (int i = tid; i < 64 * 32; i += 256) {
            int r = i >> 5, c = i & 31;
            As[r][c] = A[(size_t)(m0 + r) * K + k0 + c];
        }
        // B tile: pair-interleave k; global reads coalesced per row, LDS stores bank 2c
        for (int i = tid; i < 16 * 128; i += 256) {
            int kp = i >> 7, c = i & 127;
            Bs[kp][c][0] = W[(size_t)(k0 + 2 * kp) * N + n0 + c];
            Bs[kp][c][1] = W[(size_t)(k0 + 2 * kp + 1) * N + n0 + c];
        }
        // prefetch next K tile into cache while we compute this one
        if (k0 + 32 < K) {
            if (tid < 64) {
                __builtin_prefetch(&A[(size_t)(m0 + tid) * K + k0 + 32], 0, 1);
            } else if (tid < 192) {
                int r = (tid - 64) >> 2, s = (tid - 64) & 3;
                __builtin_prefetch(&W[(size_t)(k0 + 32 + r) * N + n0 + s * 32], 0, 1);
            }
        }
        __syncthreads();

        // fragment double-buffer: load t+1 before issuing t's WMMAs
        v2f a0c = *(const v2f*)&As[rowA0][half * 2];
        v2f a1c = *(const v2f*)&As[rowA1][half * 2];
        v2f b0c = *(const v2f*)&Bs[half][colB0][0];
        v2f b1c = *(const v2f*)&Bs[half][colB1][0];
#pragma unroll
        for (int t = 0; t < 8; ++t) {
            const int kkn = ((t + 1) & 7) * 4 + half * 2;
            v2f a0n = *(const v2f*)&As[rowA0][kkn];
            v2f a1n = *(const v2f*)&As[rowA1][kkn];
            v2f b0n = *(const v2f*)&Bs[kkn >> 1][colB0][0];
            v2f b1n = *(const v2f*)&Bs[kkn >> 1][colB1][0];
            acc[0][0] = wmma4(a0c, b0c, acc[0][0]);
            acc[0][1] = wmma4(a0c, b1c, acc[0][1]);
            acc[1][0] = wmma4(a1c, b0c, acc[1][0]);
            acc[1][1] = wmma4(a1c, b1c, acc[1][1]);
            a0c = a0n; a1c = a1n; b0c = b0n; b1c = b1n;
        }
        __syncthreads();
    }

#pragma unroll
    for (int im = 0; im < 2; ++im) {
#pragma unroll
        for (int jn = 0; jn < 2; ++jn) {
            const int col = n0 + wn * 32 + jn * 16 + lane16;
            const float bv = bias ? bias[col] : 0.0f;
#pragma unroll
            for (int vi = 0; vi < 8; ++vi) {
                const int row = m0 + wm * 32 + im * 16 + vi + half * 8;
                float val = acc[im][jn][vi] + bv;
                if (relu) val = fmaxf(val, 0.0f);
                if (mfv) val *= mfv[row];
                if (res) val += res[(size_t)row * N + col];
                C[(size_t)row * N + col] = val;
            }
        }
    }
}

// -------------------------------------------------------------------------
// Flash-style attention, per (b,h): Q[S,32] K[S,32] V[S,32] -> O[S,32]
// One block = 64 query rows x one (b,h). 128 threads = 4 waves; each wave
// owns 16 query rows. Online softmax; key mask as -1e9 bias.
// K/V chunks stored pair-interleaved for single-b64 B-operand fragments.
// grid = (S/64, B*H)
// -------------------------------------------------------------------------
__global__ void attn_kernel(const float* __restrict__ q, const float* __restrict__ k,
                            const float* __restrict__ v, const float* __restrict__ mfv,
                            float* __restrict__ o) {
    __shared__ __align__(16) float Qs[64][34];     // [row][d]
    __shared__ __align__(16) float Ks[16][80][2];  // [d/2][key][d&1]   (kp stride 160 == 32 mod 64)
    __shared__ __align__(16) float Vs[32][48][2];  // [key/2][d][key&1] (kp stride 96  == 32 mod 64)
    __shared__ __align__(16) float Ps[64][68];     // scores / probs
    __shared__ float mb[64];
    __shared__ float mrun[64], lrun[64], alph[64];

    const int S = 2048, LD = 256;
    const int tid = threadIdx.x;   // 128
    const int lane = tid & 31;
    const int wave = tid >> 5;
    const int half = lane >> 4;
    const int lane16 = lane & 15;
    const int bh = blockIdx.y;
    const int b = bh >> 3;
    const int h = bh & 7;
    const int q0 = blockIdx.x * 64;
    const int qrow = wave * 16 + lane16;

    const size_t baseQ = ((size_t)b * S + q0) * LD + h * 32;
    for (int i = tid; i < 64 * 32; i += 128) {
        int r = i >> 5, c = i & 31;
        Qs[r][c] = q[baseQ + (size_t)r * LD + c];
    }
    if (tid < 64) { mrun[tid] = -3.0e38f; lrun[tid] = 0.0f; }
    v8f oacc[2] = {};
    __syncthreads();

    for (int j0 = 0; j0 < S; j0 += 64) {
        const size_t baseK = ((size_t)b * S + j0) * LD + h * 32;
        // K: pair-interleave the d dimension (one b64 per lane, conflict-free stores)
        for (int i = tid; i < 16 * 64; i += 128) {
            int r = i & 63, kp = i >> 6;
            Ks[kp][r][0] = k[baseK + (size_t)r * LD + 2 * kp];
            Ks[kp][r][1] = k[baseK + (size_t)r * LD + 2 * kp + 1];
        }
        // V: pair-interleave the key dimension (coalesced global, conflict-free stores)
        for (int i = tid; i < 32 * 32; i += 128) {
            int n = i & 31, kp = i >> 5;
            Vs[kp][n][0] = v[baseK + (size_t)(2 * kp) * LD + n];
            Vs[kp][n][1] = v[baseK + (size_t)(2 * kp + 1) * LD + n];
        }
        if (tid < 64) mb[tid] = (mfv[(size_t)b * S + j0 + tid] != 0.0f) ? 0.0f : -1e9f;
        // prefetch next chunk
        if (j0 + 64 < S) {
            const size_t baseN = ((size_t)b * S + j0 + 64) * LD + h * 32;
            if (tid < 64) __builtin_prefetch(&k[baseN + (size_t)tid * LD], 0, 1);
            else          __builtin_prefetch(&v[baseN + (size_t)(tid - 64) * LD], 0, 1);
        }
        __syncthreads();

        // ---- scores: 16x64 strip per wave,  S = Q(16x32) * K^T(32x64) ----
        v8f sacc[4] = {};
        v2f ac = *(const v2f*)&Qs[qrow][half * 2];
        v2f bc0 = *(const v2f*)&Ks[half][0 * 16 + lane16][0];
        v2f bc1 = *(const v2f*)&Ks[half][1 * 16 + lane16][0];
        v2f bc2 = *(const v2f*)&Ks[half][2 * 16 + lane16][0];
        v2f bc3 = *(const v2f*)&Ks[half][3 * 16 + lane16][0];
#pragma unroll
        for (int t = 0; t < 8; ++t) {
            const int kkn = ((t + 1) & 7) * 4 + half * 2;
            v2f an = *(const v2f*)&Qs[qrow][kkn];
            v2f bn0 = *(const v2f*)&Ks[kkn >> 1][0 * 16 + lane16][0];
            v2f bn1 = *(const v2f*)&Ks[kkn >> 1][1 * 16 + lane16][0];
            v2f bn2 = *(const v2f*)&Ks[kkn >> 1][2 * 16 + lane16][0];
            v2f bn3 = *(const v2f*)&Ks[kkn >> 1][3 * 16 + lane16][0];
            sacc[0] = wmma4(ac, bc0, sacc[0]);
            sacc[1] = wmma4(ac, bc1, sacc[1]);
            sacc[2] = wmma4(ac, bc2, sacc[2]);
            sacc[3] = wmma4(ac, bc3, sacc[3]);
            ac = an; bc0 = bn0; bc1 = bn1; bc2 = bn2; bc3 = bn3;
        }
        const float inv = 0.17677669529663687f;   // 1/sqrt(32)
#pragma unroll
        for (int nt = 0; nt < 4; ++nt) {
            const int col = nt * 16 + lane16;
#pragma unroll
            for (int vi = 0; vi < 8; ++vi) {
                const int row = wave * 16 + vi + half * 8;
                Ps[row][col] = sacc[nt][vi] * inv + mb[col];
            }
        }
        __syncthreads();

        // ---- online softmax per query row ----
        if (tid < 64) {
            float mo = mrun[tid];
            float mc = mo;
            for (int j = 0; j < 64; ++j) mc = fmaxf(mc, Ps[tid][j]);
            float al = __expf(mo - mc);
            float ls = 0.0f;
            for (int j = 0; j < 64; ++j) {
                float p = __expf(Ps[tid][j] - mc);
                Ps[tid][j] = p;
                ls += p;
            }
            lrun[tid] = lrun[tid] * al + ls;
            mrun[tid] = mc;
            alph[tid] = al;
        }
        __syncthreads();

        // ---- rescale running O, then O += P(16x64) * V(64x32) ----
        float av[8];
#pragma unroll
        for (int vi = 0; vi < 8; ++vi) av[vi] = alph[wave * 16 + vi + half * 8];
#pragma unroll
        for (int nt = 0; nt < 2; ++nt)
#pragma unroll
            for (int vi = 0; vi < 8; ++vi) oacc[nt][vi] *= av[vi];

        v2f pc = *(const v2f*)&Ps[qrow][half * 2];
        v2f vc0 = *(const v2f*)&Vs[half][0 * 16 + lane16][0];
        v2f vc1 = *(const v2f*)&Vs[half][1 * 16 + lane16][0];
#pragma unroll
        for (int t = 0; t < 16; ++t) {
            const int kkn = ((t + 1) & 15) * 4 + half * 2;
            v2f pn = *(const v2f*)&Ps[qrow][kkn];
            v2f vn0 = *(const v2f*)&Vs[kkn >> 1][0 * 16 + lane16][0];
            v2f vn1 = *(const v2f*)&Vs[kkn >> 1][1 * 16 + lane16][0];
            oacc[0] = wmma4(pc, vc0, oacc[0]);
            oacc[1] = wmma4(pc, vc1, oacc[1]);
            pc = pn; vc0 = vn0; vc1 = vn1;
        }
        __syncthreads();
    }

    float linv[8];
#pragma unroll
    for (int vi = 0; vi < 8; ++vi) linv[vi] = 1.0f / lrun[wave * 16 + vi + half * 8];
#pragma unroll
    for (int nt = 0; nt < 2; ++nt) {
        const int col = nt * 16 + lane16;
#pragma unroll
        for (int vi = 0; vi < 8; ++vi) {
            const int row = wave * 16 + vi + half * 8;
            o[((size_t)b * S + q0 + row) * LD + h * 32 + col] = oacc[nt][vi] * linv[vi];
        }
    }
}

// -------------------------------------------------------------------------
extern "C" void kernel_launch(void* const* d_in, const int* in_sizes, int n_in,
                              void* d_out, int out_size, void* d_ws, size_t ws_size,
                              hipStream_t stream) {
    (void)in_sizes; (void)n_in; (void)out_size; (void)ws_size;
    const int B = 4, S = 2048, D = 256, F = 1024, L = 4;
    const int M = B * S;  // 8192

    const float* x_in  = (const float*)d_in[0];
    const int*   mask  = (const int*)d_in[1];
    const float* w_qs  = (const float*)d_in[2];
    const float* w_ks  = (const float*)d_in[3];
    const float* w_vs  = (const float*)d_in[4];
    const float* w_fc  = (const float*)d_in[5];
    const float* ln1_g = (const float*)d_in[6];
    const float* ln1_b = (const float*)d_in[7];
    const float* w1    = (const float*)d_in[8];
    const float* b1    = (const float*)d_in[9];
    const float* w2    = (const float*)d_in[10];
    const float* b2    = (const float*)d_in[11];
    const float* ln2_g = (const float*)d_in[12];
    const float* ln2_b = (const float*)d_in[13];

    float* ws    = (float*)d_ws;
    float* x_cur = ws;                         // [M,D]
    float* xn    = x_cur + (size_t)M * D;      // [M,D]  (reused as x2 residual)
    float* qb    = xn + (size_t)M * D;         // [M,D]  (reused for y)
    float* kb    = qb + (size_t)M * D;
    float* vb    = kb + (size_t)M * D;
    float* ob    = vb + (size_t)M * D;
    float* hb    = ob + (size_t)M * D;         // [M,F]
    float* mfv   = hb + (size_t)M * F;         // [M]

    hipMemcpyAsync(x_cur, x_in, sizeof(float) * (size_t)M * D,
                   hipMemcpyDeviceToDevice, stream);
    mf_kernel<<<M / 256, 256, 0, stream>>>(mask, mfv, M);

    const dim3 gD(M / 64, D / 128);
    const dim3 gF(M / 64, F / 128);
    const dim3 gA(S / 64, B * 8);

    for (int l = 0; l < L; ++l) {
        const float* wq = w_qs + (size_t)l * D * D;
        const float* wk = w_ks + (size_t)l * D * D;
        const float* wv = w_vs + (size_t)l * D * D;
        const float* wf = w_fc + (size_t)l * D * D;
        const float* w1l = w1 + (size_t)l * D * F;
        const float* w2l = w2 + (size_t)l * F * D;
        const float* b1l = b1 + (size_t)l * F;
        const float* b2l = b2 + (size_t)l * D;

        ln_kernel<<<M, 256, 0, stream>>>(x_cur, xn, ln1_g + (size_t)l * D, ln1_b + (size_t)l * D);
        gemm_kernel<<<gD, 256, 0, stream>>>(xn, wq, qb, nullptr, mfv, nullptr, D, D, 0);
        gemm_kernel<<<gD, 256, 0, stream>>>(xn, wk, kb, nullptr, mfv, nullptr, D, D, 0);
        gemm_kernel<<<gD, 256, 0, stream>>>(xn, wv, vb, nullptr, mfv, nullptr, D, D, 0);
        attn_kernel<<<gA, 128, 0, stream>>>(qb, kb, vb, mfv, ob);
        gemm_kernel<<<gD, 256, 0, stream>>>(ob, wf, xn, nullptr, mfv, x_cur, D, D, 0);
        gemm_kernel<<<gF, 256, 0, stream>>>(xn, w1l, hb, b1l, nullptr, nullptr, D, F, 1);
        gemm_kernel<<<gD, 256, 0, stream>>>(hb, w2l, qb, b2l, nullptr, xn, F, D, 0);
        ln_kernel<<<M, 256, 0, stream>>>(qb, x_cur, ln2_g + (size_t)l * D, ln2_b + (size_t)l * D);
    }

    hipMemcpyAsync(d_out, x_cur, sizeof(float) * (size_t)M * D,
                   hipMemcpyDeviceToDevice, stream);
}